// GCN_encoder_14130442403926
// MI455X (gfx1250) — compile-verified
//
#include <hip/hip_runtime.h>

typedef float v2f __attribute__((ext_vector_type(2)));
typedef float v8f __attribute__((ext_vector_type(8)));

#define N_NODES 100000
#define N_EDGES 1600000
#define HID 64

// ---------------- utility kernels ----------------

__global__ void zero_kernel(float* __restrict__ p, int n) {
    int i = blockIdx.x * blockDim.x + threadIdx.x;
    if (i < n) p[i] = 0.0f;
}

__global__ void degree_kernel(const int* __restrict__ src, const int* __restrict__ dst,
                              float* __restrict__ deg_out, float* __restrict__ deg_in, int E) {
    int e = blockIdx.x * blockDim.x + threadIdx.x;
    if (e < E) {
        atomicAdd(&deg_out[src[e]], 1.0f);
        atomicAdd(&deg_in[dst[e]], 1.0f);
    }
}

// in-place: deg -> (deg>0 ? deg : 1)^-1/2
__global__ void norm_kernel(float* __restrict__ deg, int n) {
    int i = blockIdx.x * blockDim.x + threadIdx.x;
    if (i < n) {
        float d = deg[i];
        d = d > 0.0f ? d : 1.0f;
        deg[i] = 1.0f / sqrtf(d);
    }
}

// ---------------- WMMA GEMM: out[M,64] = (A * nsrc[:,None]) @ W[K,64] ----------------
// One workgroup (128 threads = 4 wave32) computes a 16x64 output tile.
// Each wave computes one 16x16 tile via V_WMMA_F32_16X16X4_F32, stepping K by 4.
//
// VGPR layouts (CDNA5 ISA 7.12.2):
//  A 16x4 f32 : lane L(<16) holds A[M=L][k], A[M=L][k+1] in v0,v1; lane L+16 holds k+2,k+3
//  B 4x16 f32 : v0 = row k (lanes 0-15, N=0..15) / row k+2 (lanes 16-31); v1 = k+1 / k+3
//  C 16x16 f32: VGPR r = row r (lanes 0-15) / row r+8 (lanes 16-31), N = lane%16
template <int K>
__global__ __launch_bounds__(128) void gemm_norm_wmma(
        const float* __restrict__ A,     // [M, K]
        const float* __restrict__ W,     // [K, 64]
        const float* __restrict__ nsrc,  // [M]
        float* __restrict__ out)         // [M, 64]
{
    const int lane   = threadIdx.x & 31;
    const int wave   = threadIdx.x >> 5;        // 0..3 -> N tile
    const int rowBase = blockIdx.x * 16;        // M tile (100000 % 16 == 0)
    const int nBase   = wave * 16;
    const int m      = rowBase + (lane & 15);
    const int khalf  = (lane >> 4) * 2;         // 0 or 2
    const int n      = nBase + (lane & 15);

    const float ns = nsrc[m];
    const float* arow = A + (size_t)m * K;

    v8f c = {};
#pragma unroll 4
    for (int k = 0; k < K; k += 4) {
        v2f a, b;
        const float* ap = arow + k + khalf;
        a.x = ap[0] * ns;
        a.y = ap[1] * ns;
        const float* bp = W + (size_t)(k + khalf) * HID + n;
        b.x = bp[0];
        b.y = bp[HID];
        c = __builtin_amdgcn_wmma_f32_16x16x4_f32(
                /*neg_a=*/false, a, /*neg_b=*/false, b,
                /*c_mod=*/(short)0, c, /*reuse_a=*/false, /*reuse_b=*/false);
    }

    const int rOff = (lane >> 4) * 8;
    float* op = out + (size_t)rowBase * HID + n;
#pragma unroll
    for (int r = 0; r < 8; ++r) {
        op[(size_t)(r + rOff) * HID] = c[r];
    }
}

// ---------------- edge scatter: agg[dst] += tmp[src] ----------------
// block = (64 channels, 4 edges); each wave32 moves 128B of one edge's row.
__global__ __launch_bounds__(256) void scatter_kernel(
        const float* __restrict__ tmp, const int* __restrict__ src,
        const int* __restrict__ dst, float* __restrict__ agg, int E)
{
    int e = blockIdx.x * blockDim.y + threadIdx.y;
    int c = threadIdx.x;   // 0..63
    if (e < E) {
        int s = src[e];
        int d = dst[e];
        float v = tmp[(size_t)s * HID + c];
        atomicAdd(&agg[(size_t)d * HID + c], v);
    }
}

// ---------------- finalize: out = relu(agg * ndst + b) ----------------
__global__ void finalize_kernel(const float* __restrict__ agg, const float* __restrict__ ndst,
                                const float* __restrict__ bias, float* __restrict__ out, int n64)
{
    int i = blockIdx.x * blockDim.x + threadIdx.x;
    if (i < n64) {
        int node = i >> 6;   // /64
        int c    = i & 63;
        float v = agg[i] * ndst[node] + bias[c];
        out[i] = v > 0.0f ? v : 0.0f;
    }
}

// ---------------- host launcher ----------------
extern "C" void kernel_launch(void* const* d_in, const int* in_sizes, int n_in,
                              void* d_out, int out_size, void* d_ws, size_t ws_size,
                              hipStream_t stream) {
    const float* features = (const float*)d_in[0];
    const float* W0 = (const float*)d_in[1];
    const float* b0 = (const float*)d_in[2];
    const float* W1 = (const float*)d_in[3];
    const float* b1 = (const float*)d_in[4];
    const float* W2 = (const float*)d_in[5];
    const float* b2 = (const float*)d_in[6];
    const int*   src = (const int*)d_in[7];
    const int*   dst = (const int*)d_in[8];

    float* ws   = (float*)d_ws;
    float* nsrc = ws;                                  // [N]
    float* ndst = ws + N_NODES;                        // [N]
    float* tmp  = ws + 2 * N_NODES;                    // [N,64]
    float* agg  = tmp + (size_t)N_NODES * HID;         // [N,64]
    float* hbuf = agg + (size_t)N_NODES * HID;         // [N,64]

    const int NH = N_NODES * HID;   // 6,400,000

    // Degrees -> norms (nsrc/ndst are adjacent: zero & rsqrt both in one pass)
    zero_kernel<<<(2 * N_NODES + 255) / 256, 256, 0, stream>>>(nsrc, 2 * N_NODES);
    degree_kernel<<<(N_EDGES + 255) / 256, 256, 0, stream>>>(src, dst, nsrc, ndst, N_EDGES);
    norm_kernel<<<(2 * N_NODES + 255) / 256, 256, 0, stream>>>(nsrc, 2 * N_NODES);

    dim3 sblk(64, 4);
    int  sgrd = (N_EDGES + 3) / 4;
    int  fgrd = (NH + 255) / 256;

    // ---- layer 0: [N,128] -> [N,64]
    gemm_norm_wmma<128><<<N_NODES / 16, 128, 0, stream>>>(features, W0, nsrc, tmp);
    zero_kernel<<<fgrd, 256, 0, stream>>>(agg, NH);
    scatter_kernel<<<sgrd, sblk, 0, stream>>>(tmp, src, dst, agg, N_EDGES);
    finalize_kernel<<<fgrd, 256, 0, stream>>>(agg, ndst, b0, hbuf, NH);

    // ---- layer 1: [N,64] -> [N,64]
    gemm_norm_wmma<64><<<N_NODES / 16, 128, 0, stream>>>(hbuf, W1, nsrc, tmp);
    zero_kernel<<<fgrd, 256, 0, stream>>>(agg, NH);
    scatter_kernel<<<sgrd, sblk, 0, stream>>>(tmp, src, dst, agg, N_EDGES);
    finalize_kernel<<<fgrd, 256, 0, stream>>>(agg, ndst, b1, hbuf, NH);

    // ---- layer 2: [N,64] -> [N,64] -> d_out
    gemm_norm_wmma<64><<<N_NODES / 16, 128, 0, stream>>>(hbuf, W2, nsrc, tmp);
    zero_kernel<<<fgrd, 256, 0, stream>>>(agg, NH);
    scatter_kernel<<<sgrd, sblk, 0, stream>>>(tmp, src, dst, agg, N_EDGES);
    finalize_kernel<<<fgrd, 256, 0, stream>>>(agg, ndst, b2, (float*)d_out, NH);
}